// SPD2_16750372454765
// MI455X (gfx1250) — compile-verified
//
#include <hip/hip_runtime.h>
#include <hip/hip_bf16.h>
#include <math.h>

// ---------------------------------------------------------------------------
// SPD step for MI455X (gfx1250).  Dense GEMMs on V_WMMA_F32_16X16X4_F32
// (fp32 in/out -> reference precision).  A-operand strips staged via
// GLOBAL_LOAD_ASYNC_TO_LDS_B128 (ASYNCcnt) and shared by 4 waves per block.
// KNN + per-point transformer on VALU/LDS.  wave32 everywhere.
// ---------------------------------------------------------------------------

#define NPTS     8192
#define CCH      128
#define NS       16
#define SEGPTS   4096
#define NCHILD   (NPTS * 2)
#define EPSV     1e-5f
#define MAXK     256
#define APAD     4          // LDS row padding (floats) -> conflict-free ds_load

typedef __attribute__((ext_vector_type(2))) float v2f;
typedef __attribute__((ext_vector_type(8))) float v8f;

// ---------------------------------------------------------------------------
// fp32 WMMA GEMM:  C[M x N] = A[M x K] * B[N x K]^T  (+ fused epilogue)
//   epilogue: (+bias[n]) -> (BN affine over n) -> (relu|tanh) -> (+add[m,n])
// Block = 128 threads (4 waves).  The block's 16-row A strip is staged once
// into LDS with async global->LDS copies; each wave computes one 16x16 C tile
// (4 consecutive N tiles per block).  A lane layout (ISA 32-bit 16x4 A):
// row = lane%16, k-pair = 2*(lane/16).  B addressed identically on row-major
// W[N][K] => B[k][n] = W[n][k].  M multiple of 16 (all call sites); N guarded.
// K must be a multiple of 4 and <= MAXK (64/128/256 here).
// ---------------------------------------------------------------------------
__global__ void __launch_bounds__(128)
gemm16_wmma(const float* __restrict__ A, const float* __restrict__ B,
            float* __restrict__ C, int M, int N, int K,
            const float* __restrict__ bias,
            const float* __restrict__ bng, const float* __restrict__ bnb,
            const float* __restrict__ bnm, const float* __restrict__ bnv,
            const float* __restrict__ addsrc, int act /*0 none,1 relu,2 tanh*/)
{
    __shared__ float sA[16 * (MAXK + APAD)];

    const int tid  = threadIdx.x;
    const int wave = tid >> 5;
    const int lane = tid & 31;
    const int half = lane >> 4;        // 0/1 -> k-pair select
    const int lr   = lane & 15;
    const int m0   = blockIdx.y << 4;
    const int n0   = (blockIdx.x * 4 + wave) << 4;
    const int S    = K + APAD;         // LDS row stride (floats)

    // ---- stage A[m0..m0+15][0..K) into LDS via async global->LDS (16B/lane)
    {
        const int kc     = K >> 2;               // 16B chunks per row
        const int chunks = 16 * kc;              // multiple of 128
        for (int c = tid; c < chunks; c += 128) {
            const int row  = c / kc;
            const int col4 = (c - row * kc) << 2;
            const unsigned long long gsrc =
                (unsigned long long)(uintptr_t)(A + (size_t)(m0 + row) * K + col4);
            const unsigned int ldst =
                (unsigned int)(uintptr_t)(&sA[row * S + col4]);
            asm volatile("global_load_async_to_lds_b128 %0, %1, off"
                         :: "v"(ldst), "v"(gsrc) : "memory");
        }
        asm volatile("s_wait_asynccnt 0x0" ::: "memory");
        __syncthreads();
    }

    const int   nrow   = n0 + lr;
    const bool  nvalid = (nrow < N);
    const int   nsafe  = nvalid ? nrow : 0;
    const float* __restrict__ brow = B + (size_t)nsafe * K;
    const float* arow = &sA[lr * S];

    v8f acc = {};
#pragma unroll 4
    for (int k = 0; k < K; k += 4) {
        const int kk = k + (half << 1);
        v2f a = *(const v2f*)(arow + kk);          // ds_load_b64, conflict-free
        v2f bv = *(const v2f*)(brow + kk);
        v2f b;
        b.x = nvalid ? bv.x : 0.0f;
        b.y = nvalid ? bv.y : 0.0f;
        acc = __builtin_amdgcn_wmma_f32_16x16x4_f32(
            /*neg_a=*/false, a, /*neg_b=*/false, b,
            /*c_mod=*/(short)0, acc, /*reuse_a=*/false, /*reuse_b=*/false);
    }

    // per-column epilogue constants (uniform across the 8 row-results)
    float bsv = bias ? bias[nsafe] : 0.0f;
    float sc = 1.0f, sh = 0.0f;
    if (bng) {
        const float inv = rsqrtf(bnv[nsafe] + EPSV);
        sc = bng[nsafe] * inv;
        sh = bnb[nsafe] - bnm[nsafe] * sc;
    }

#pragma unroll
    for (int r = 0; r < 8; ++r) {
        const int m = m0 + r + (half << 3);
        float v = acc[r] + bsv;
        if (bng) v = v * sc + sh;
        if (act == 1)      v = fmaxf(v, 0.0f);
        else if (act == 2) v = tanhf(v);
        if (nvalid) {
            if (addsrc) v += addsrc[(size_t)m * N + nrow];
            C[(size_t)m * N + nrow] = v;
        }
    }
}

// ---------------------------------------------------------------------------
// Brute-force segment-local KNN (NS=16 nearest incl. self).
// 1 thread per query point; segment points streamed through LDS tiles.
// Register top-16 kept with a fully unrolled compare-swap chain (stable wrt
// index order -> matches jax.lax.top_k tie behavior).
// ---------------------------------------------------------------------------
__global__ void __launch_bounds__(256)
knn_kernel(const float* __restrict__ pcd, int* __restrict__ gidx)
{
    const int TILE = 512;
    __shared__ float sx[TILE], sy[TILE], sz[TILE];

    const int i    = blockIdx.x * 256 + threadIdx.x;   // query point
    const int seg  = i >> 12;                          // /4096
    const int base = seg << 12;

    const float qx = pcd[i * 3 + 0];
    const float qy = pcd[i * 3 + 1];
    const float qz = pcd[i * 3 + 2];

    float bd[NS];
    int   bi[NS];
#pragma unroll
    for (int s = 0; s < NS; ++s) { bd[s] = 3.4e38f; bi[s] = 0; }

    for (int t0 = 0; t0 < SEGPTS; t0 += TILE) {
        __syncthreads();
        for (int l = threadIdx.x; l < TILE; l += 256) {
            const int p = base + t0 + l;
            sx[l] = pcd[p * 3 + 0];
            sy[l] = pcd[p * 3 + 1];
            sz[l] = pcd[p * 3 + 2];
        }
        __syncthreads();
        for (int l = 0; l < TILE; ++l) {
            const float dx = qx - sx[l], dy = qy - sy[l], dz = qz - sz[l];
            const float d2 = dx * dx + dy * dy + dz * dz;
            if (d2 < bd[NS - 1]) {
                float nd = d2; int ni = t0 + l;
#pragma unroll
                for (int p = 0; p < NS; ++p) {
                    if (nd < bd[p]) {
                        const float td = bd[p]; const int ti = bi[p];
                        bd[p] = nd; bi[p] = ni; nd = td; ni = ti;
                    }
                }
            }
        }
    }
#pragma unroll
    for (int s = 0; s < NS; ++s) gidx[i * NS + s] = base + bi[s];
}

// ---------------------------------------------------------------------------
// PointTransformer layer: one 128-thread block per point.
//   H[i] = relu(bn2( sum_j (xv_g + pr) * softmax_j(w) )) + K[i]
// ---------------------------------------------------------------------------
__global__ void __launch_bounds__(128)
ptlayer_kernel(const float* __restrict__ Kbuf,
               const float* __restrict__ xq, const float* __restrict__ xk,
               const float* __restrict__ xv,
               const float* __restrict__ pcd, const int* __restrict__ gidx,
               const float* __restrict__ lp0w, const float* __restrict__ lp0b,
               const float* __restrict__ lpg,  const float* __restrict__ lpb,
               const float* __restrict__ lpm,  const float* __restrict__ lpv,
               const float* __restrict__ lp1w, const float* __restrict__ lp1b,
               const float* __restrict__ w0g,  const float* __restrict__ w0b,
               const float* __restrict__ w0m,  const float* __restrict__ w0v,
               const float* __restrict__ l0w,  const float* __restrict__ l0b,
               const float* __restrict__ w1g,  const float* __restrict__ w1b,
               const float* __restrict__ w1m,  const float* __restrict__ w1v,
               const float* __restrict__ l1w,  const float* __restrict__ l1b,
               const float* __restrict__ bn2g, const float* __restrict__ bn2b,
               const float* __restrict__ bn2m, const float* __restrict__ bn2v,
               float* __restrict__ H)
{
    __shared__ int   sg[NS];
    __shared__ float pr3[NS][4];        // relu(bn(lin3x3(p_r)))
    __shared__ float prs[NS][CCH];      // lp1 output per neighbor
    __shared__ float as_[NS][CCH];      // relu(bn0(xk_g - xq + pr))
    __shared__ float w0a[NS][NS];       // relu(bn1(l0(.)))
    __shared__ float wsm[NS][NS];       // softmax weights

    const int i   = blockIdx.x;
    const int tid = threadIdx.x;

    if (tid < NS) sg[tid] = gidx[i * NS + tid];
    __syncthreads();

    // --- linear_p first stage (3->3, BN3, relu), one thread per neighbor ---
    if (tid < NS) {
        const int gj = sg[tid];
        const float px = pcd[gj * 3 + 0] - pcd[i * 3 + 0];
        const float py = pcd[gj * 3 + 1] - pcd[i * 3 + 1];
        const float pz = pcd[gj * 3 + 2] - pcd[i * 3 + 2];
#pragma unroll
        for (int c = 0; c < 3; ++c) {
            float t = lp0w[c * 3 + 0] * px + lp0w[c * 3 + 1] * py +
                      lp0w[c * 3 + 2] * pz + lp0b[c];
            const float s = lpg[c] * rsqrtf(lpv[c] + EPSV);
            t = (t - lpm[c]) * s + lpb[c];
            pr3[tid][c] = fmaxf(t, 0.0f);
        }
    }
    __syncthreads();

    // --- lp1 (3->128), w_in = xk_g - xq + pr, a = relu(bn0(w_in)) ----------
    {
        const int t = tid;  // channel
        const float w0c = lp1w[t * 3 + 0], w1c = lp1w[t * 3 + 1],
                    w2c = lp1w[t * 3 + 2], bc = lp1b[t];
        const float xqv = xq[(size_t)i * CCH + t];
        const float s0  = w0g[t] * rsqrtf(w0v[t] + EPSV);
        const float sh0 = w0b[t] - w0m[t] * s0;
#pragma unroll
        for (int j = 0; j < NS; ++j) {
            const float pr = w0c * pr3[j][0] + w1c * pr3[j][1] +
                             w2c * pr3[j][2] + bc;
            prs[j][t] = pr;
            const float win = xk[(size_t)sg[j] * CCH + t] - xqv + pr;
            as_[j][t] = fmaxf(win * s0 + sh0, 0.0f);
        }
    }
    __syncthreads();

    // --- l0: 128 -> 16 per neighbor, then bn1+relu --------------------------
#pragma unroll
    for (int rep = 0; rep < 2; ++rep) {
        const int p = tid + rep * 128;   // 256 outputs: (j, c)
        const int j = p >> 4, c = p & 15;
        float acc = l0b[c];
        const float* __restrict__ wr = l0w + c * CCH;
        for (int t = 0; t < CCH; ++t) acc += as_[j][t] * wr[t];
        const float s1 = w1g[c] * rsqrtf(w1v[c] + EPSV);
        acc = (acc - w1m[c]) * s1 + w1b[c];
        w0a[j][c] = fmaxf(acc, 0.0f);
    }
    __syncthreads();

    // --- l1: 16 -> 16 per neighbor -----------------------------------------
#pragma unroll
    for (int rep = 0; rep < 2; ++rep) {
        const int p = tid + rep * 128;
        const int j = p >> 4, c = p & 15;
        float acc = l1b[c];
        const float* __restrict__ wr = l1w + c * NS;
#pragma unroll
        for (int k = 0; k < NS; ++k) acc += w0a[j][k] * wr[k];
        wsm[j][c] = acc;
    }
    __syncthreads();

    // --- softmax over the 16 neighbors, per shared channel -----------------
    if (tid < NS) {
        const int c = tid;
        float mx = -3.4e38f;
#pragma unroll
        for (int j = 0; j < NS; ++j) mx = fmaxf(mx, wsm[j][c]);
        float e[NS]; float sum = 0.0f;
#pragma unroll
        for (int j = 0; j < NS; ++j) { e[j] = __expf(wsm[j][c] - mx); sum += e[j]; }
        const float inv = 1.0f / sum;
#pragma unroll
        for (int j = 0; j < NS; ++j) wsm[j][c] = e[j] * inv;
    }
    __syncthreads();

    // --- aggregate + bn2 + relu + identity ---------------------------------
    {
        const int t = tid;
        float h = 0.0f;
#pragma unroll
        for (int j = 0; j < NS; ++j)
            h += (xv[(size_t)sg[j] * CCH + t] + prs[j][t]) * wsm[j][t & 15];
        const float s2 = bn2g[t] * rsqrtf(bn2v[t] + EPSV);
        h = (h - bn2m[t]) * s2 + bn2b[t];
        h = fmaxf(h, 0.0f) + Kbuf[(size_t)i * CCH + t];
        H[(size_t)i * CCH + t] = h;
    }
}

// ---------------------------------------------------------------------------
// ConvTranspose1d(32,128,k=2,s=2) + concat with repeated H -> cat (2N x 256)
// ---------------------------------------------------------------------------
__global__ void __launch_bounds__(256)
cat_kernel(const float* __restrict__ fc, const float* __restrict__ psw,
           const float* __restrict__ H, float* __restrict__ cat)
{
    const int idx = blockIdx.x * 256 + threadIdx.x;   // NCHILD*128
    const int row = idx >> 7;
    const int o   = idx & 127;
    const int n   = row >> 1, j = row & 1;
    float acc = 0.0f;
#pragma unroll 8
    for (int c = 0; c < 32; ++c)
        acc += fc[n * 32 + c] * psw[c * 256 + o * 2 + j];
    cat[(size_t)row * 256 + o]       = acc;
    cat[(size_t)row * 256 + 128 + o] = H[(size_t)n * CCH + o];
}

// ---------------------------------------------------------------------------
// pcd_child = repeat(pcd_prev, 2) + delta
// ---------------------------------------------------------------------------
__global__ void __launch_bounds__(256)
child_kernel(const float* __restrict__ pcd, const float* __restrict__ delta,
             float* __restrict__ out)
{
    const int idx = blockIdx.x * 256 + threadIdx.x;
    if (idx < NCHILD * 3) {
        const int i = idx / 3, d = idx - i * 3;
        out[idx] = pcd[(i >> 1) * 3 + d] + delta[idx];
    }
}

// ---------------------------------------------------------------------------
// Host-side orchestration
// ---------------------------------------------------------------------------
static inline void launch_gemm(hipStream_t st, const float* A, const float* B,
                               float* C, int M, int N, int K,
                               const float* bias,
                               const float* bg, const float* bb,
                               const float* bm, const float* bv,
                               const float* add, int act)
{
    dim3 grid((N + 63) / 64, M / 16);
    gemm16_wmma<<<grid, 128, 0, st>>>(A, B, C, M, N, K, bias, bg, bb, bm, bv, add, act);
}

extern "C" void kernel_launch(void* const* d_in, const int* in_sizes, int n_in,
                              void* d_out, int out_size, void* d_ws, size_t ws_size,
                              hipStream_t stream)
{
    // ---- input map (recursive insertion-order flattening of setup_inputs) ----
    const float* pcd    = (const float*)d_in[0];   // (8192,3)
    const float* K_prev = (const float*)d_in[1];   // (8192,128)
    /* d_in[2] = o (int64[2]) -- segments hardcoded */
    const float* l1w = (const float*)d_in[3],  *l1b = (const float*)d_in[4];
    const float* b1g = (const float*)d_in[5],  *b1b = (const float*)d_in[6];
    const float* b1m = (const float*)d_in[7],  *b1v = (const float*)d_in[8];
    const float* b2g = (const float*)d_in[9],  *b2b = (const float*)d_in[10];
    const float* b2m = (const float*)d_in[11], *b2v = (const float*)d_in[12];
    const float* lqw = (const float*)d_in[13], *lqb = (const float*)d_in[14];
    const float* lkw = (const float*)d_in[15], *lkb = (const float*)d_in[16];
    const float* lvw = (const float*)d_in[17], *lvb = (const float*)d_in[18];
    const float* lp0w = (const float*)d_in[19], *lp0b = (const float*)d_in[20];
    const float* lpg = (const float*)d_in[21], *lpb = (const float*)d_in[22];
    const float* lpm = (const float*)d_in[23], *lpv = (const float*)d_in[24];
    const float* lp1w = (const float*)d_in[25], *lp1b = (const float*)d_in[26];
    const float* w0g = (const float*)d_in[27], *w0b = (const float*)d_in[28];
    const float* w0m = (const float*)d_in[29], *w0v = (const float*)d_in[30];
    const float* l0w = (const float*)d_in[31], *l0b = (const float*)d_in[32];
    const float* w1g = (const float*)d_in[33], *w1b = (const float*)d_in[34];
    const float* w1m = (const float*)d_in[35], *w1v = (const float*)d_in[36];
    const float* lw1w = (const float*)d_in[37], *lw1b = (const float*)d_in[38];
    const float* ps0w = (const float*)d_in[39], *ps0b = (const float*)d_in[40];
    const float* ps1w = (const float*)d_in[41], *ps1b = (const float*)d_in[42];
    const float* psw  = (const float*)d_in[43];                       // (32,128,2)
    const float* scw  = (const float*)d_in[44], *scb = (const float*)d_in[45];
    const float* r1w  = (const float*)d_in[46], *r1b = (const float*)d_in[47];
    const float* r2w  = (const float*)d_in[48], *r2b = (const float*)d_in[49];
    const float* d0w  = (const float*)d_in[50], *d0b = (const float*)d_in[51];
    const float* d1w  = (const float*)d_in[52], *d1b = (const float*)d_in[53];
    (void)in_sizes; (void)n_in; (void)out_size; (void)ws_size;

    // ---- workspace layout (float offsets; phases alias, peak ~34 MB) --------
    float* ws = (float*)d_ws;
    float* Kbuf  = ws + 0;               // 8192*128        (phase 1)
    float* xq    = ws + 1048576;
    float* xk    = ws + 2097152;
    float* xv    = ws + 3145728;
    float* Hbuf  = ws + 4194304;         // 8192*128
    float* t64   = ws + 5242880;         // 8192*64
    float* fc    = ws + 5767168;         // 8192*32
    float* catb  = ws + 0;               // 16384*256  (aliases Kbuf/xq/xk/xv)
    float* r1    = ws + 4194304;         // 16384*128  (aliases H/t64/fc)
    float* scb_f = ws + 6291456;         // 16384*128
    float* t64b  = ws + 0;               // 16384*64   (aliases cat)
    float* delta = ws + 1048576;         // 16384*3
    int*   gidx  = (int*)(ws + 8388608); // 8192*16 ints

    float* out_pcd = (float*)d_out;              // (16384,3)
    float* K_curr  = (float*)d_out + NCHILD * 3; // (16384,128)

    // 1) K = relu(bn1(linear1(K_prev)))
    launch_gemm(stream, K_prev, l1w, Kbuf, NPTS, CCH, CCH,
                l1b, b1g, b1b, b1m, b1v, nullptr, 1);
    // 2) xq / xk / xv
    launch_gemm(stream, Kbuf, lqw, xq, NPTS, CCH, CCH, lqb,
                nullptr, nullptr, nullptr, nullptr, nullptr, 0);
    launch_gemm(stream, Kbuf, lkw, xk, NPTS, CCH, CCH, lkb,
                nullptr, nullptr, nullptr, nullptr, nullptr, 0);
    launch_gemm(stream, Kbuf, lvw, xv, NPTS, CCH, CCH, lvb,
                nullptr, nullptr, nullptr, nullptr, nullptr, 0);
    // 3) segment-local KNN
    knn_kernel<<<NPTS / 256, 256, 0, stream>>>(pcd, gidx);
    // 4) point transformer layer -> H
    ptlayer_kernel<<<NPTS, 128, 0, stream>>>(
        Kbuf, xq, xk, xv, pcd, gidx,
        lp0w, lp0b, lpg, lpb, lpm, lpv, lp1w, lp1b,
        w0g, w0b, w0m, w0v, l0w, l0b,
        w1g, w1b, w1m, w1v, lw1w, lw1b,
        b2g, b2b, b2m, b2v, Hbuf);
    // 5) mlp_ps: H -> 64 (relu) -> 32
    launch_gemm(stream, Hbuf, ps0w, t64, NPTS, 64, CCH, ps0b,
                nullptr, nullptr, nullptr, nullptr, nullptr, 1);
    launch_gemm(stream, t64, ps1w, fc, NPTS, 32, 64, ps1b,
                nullptr, nullptr, nullptr, nullptr, nullptr, 0);
    // 6) feat_child + concat(H_up) -> cat
    cat_kernel<<<(NCHILD * CCH) / 256, 256, 0, stream>>>(fc, psw, Hbuf, catb);
    // 7) MLP_Res: K_curr = res_2(relu(res_1(cat))) + res_sc(cat)
    launch_gemm(stream, catb, r1w, r1, NCHILD, CCH, 2 * CCH, r1b,
                nullptr, nullptr, nullptr, nullptr, nullptr, 1);
    launch_gemm(stream, catb, scw, scb_f, NCHILD, CCH, 2 * CCH, scb,
                nullptr, nullptr, nullptr, nullptr, nullptr, 0);
    launch_gemm(stream, r1, r2w, K_curr, NCHILD, CCH, CCH, r2b,
                nullptr, nullptr, nullptr, nullptr, scb_f, 0);
    // 8) mlp_delta: K_curr -> 64 (relu) -> 3 (tanh)
    launch_gemm(stream, K_curr, d0w, t64b, NCHILD, 64, CCH, d0b,
                nullptr, nullptr, nullptr, nullptr, nullptr, 1);
    launch_gemm(stream, t64b, d1w, delta, NCHILD, 3, 64, d1b,
                nullptr, nullptr, nullptr, nullptr, nullptr, 2);
    // 9) pcd_child = repeat(pcd_prev,2) + delta
    child_kernel<<<(NCHILD * 3 + 255) / 256, 256, 0, stream>>>(pcd, delta, out_pcd);
}